// NoteDropout_21638045237926
// MI455X (gfx1250) — compile-verified
//
#include <hip/hip_runtime.h>

// out[i] = input[i] * (mask[i] > p ? mask[i] : 0)
//
// Roofline: 37.75M fp32 elems, 12 B/elem => ~453 MB HBM traffic, ~19.4 us
// floor at 23.3 TB/s; ~0.17 FLOP/byte => purely memory-bound, WMMA irrelevant.
// Strategy: B128 non-temporal loads/stores (CDNA5 TH=NT cache control; the
// 302 MB read stream has zero reuse and must not thrash the 192 MB L2),
// wave32 blocks of 256 threads, 4-way unrolled grid-stride loop => 8 B128
// loads in flight per wave before the first s_wait_loadcnt.
//
// Fast path uses 32-bit unsigned offsets (tensors are 151 MB each, byte
// offsets < 2^31) so the backend emits SGPR-base + 32-bit-VGPR-offset (GVS)
// addressing: one v_add_nc_u32 per pointer per trip instead of 64-bit pairs.

typedef __attribute__((ext_vector_type(4))) float v4f;

template <typename IDX>
__global__ __launch_bounds__(256) void note_dropout_kernel(
    const float* __restrict__ in,
    const float* __restrict__ p_ptr,
    const float* __restrict__ mask,
    float* __restrict__ out,
    IDX n)
{
    const float p = p_ptr[0];   // uniform scalar load (KMcnt path)

    const IDX n4     = n >> 2;
    const IDX tid    = (IDX)blockIdx.x * (IDX)blockDim.x + (IDX)threadIdx.x;
    const IDX stride = (IDX)gridDim.x * (IDX)blockDim.x;

    const v4f* __restrict__ in4 = (const v4f*)in;
    const v4f* __restrict__ mk4 = (const v4f*)mask;
    v4f*       __restrict__ o4  = (v4f*)out;

    IDX i = tid;

    // Main path: 4 x float4 per thread per trip; consecutive lanes touch
    // consecutive 16B chunks -> fully coalesced B128 transactions.
    for (; i + 3 * stride < n4; i += 4 * stride) {
        v4f a0 = __builtin_nontemporal_load(in4 + i);
        v4f a1 = __builtin_nontemporal_load(in4 + i + stride);
        v4f a2 = __builtin_nontemporal_load(in4 + i + 2 * stride);
        v4f a3 = __builtin_nontemporal_load(in4 + i + 3 * stride);
        v4f m0 = __builtin_nontemporal_load(mk4 + i);
        v4f m1 = __builtin_nontemporal_load(mk4 + i + stride);
        v4f m2 = __builtin_nontemporal_load(mk4 + i + 2 * stride);
        v4f m3 = __builtin_nontemporal_load(mk4 + i + 3 * stride);

        v4f r0, r1, r2, r3;
#pragma unroll
        for (int c = 0; c < 4; ++c) {
            r0[c] = a0[c] * (m0[c] > p ? m0[c] : 0.0f);
            r1[c] = a1[c] * (m1[c] > p ? m1[c] : 0.0f);
            r2[c] = a2[c] * (m2[c] > p ? m2[c] : 0.0f);
            r3[c] = a3[c] * (m3[c] > p ? m3[c] : 0.0f);
        }

        __builtin_nontemporal_store(r0, o4 + i);
        __builtin_nontemporal_store(r1, o4 + i + stride);
        __builtin_nontemporal_store(r2, o4 + i + 2 * stride);
        __builtin_nontemporal_store(r3, o4 + i + 3 * stride);
    }

    // Vector remainder.
    for (; i < n4; i += stride) {
        v4f a = __builtin_nontemporal_load(in4 + i);
        v4f m = __builtin_nontemporal_load(mk4 + i);
        v4f r;
#pragma unroll
        for (int c = 0; c < 4; ++c) r[c] = a[c] * (m[c] > p ? m[c] : 0.0f);
        __builtin_nontemporal_store(r, o4 + i);
    }

    // Scalar remainder (n % 4) -- empty for the reference shape.
    for (IDX j = (n4 << 2) + tid; j < n; j += stride) {
        float m = mask[j];
        out[j] = in[j] * (m > p ? m : 0.0f);
    }
}

extern "C" void kernel_launch(void* const* d_in, const int* in_sizes, int n_in,
                              void* d_out, int out_size, void* d_ws, size_t ws_size,
                              hipStream_t stream) {
    const float* in   = (const float*)d_in[0];
    const float* p    = (const float*)d_in[1];  // single-element device scalar
    const float* mask = (const float*)d_in[2];
    float*       out  = (float*)d_out;

    const long long n  = (long long)in_sizes[0];
    const long long n4 = n >> 2;

    const int       block     = 256;                        // 8 wave32 waves
    const long long unroll    = 4;
    long long       blocks_ll = (n4 + (long long)block * unroll - 1) /
                                ((long long)block * unroll);
    if (blocks_ll < 1) blocks_ll = 1;
    if (blocks_ll > 0x7FFFFFFFLL) blocks_ll = 0x7FFFFFFFLL;
    const int grid = (int)blocks_ll;                        // 9216 for ref shape

    if (n < (1LL << 28)) {
        // 32-bit offset fast path (reference shape: n = 37,748,736).
        note_dropout_kernel<unsigned int><<<grid, block, 0, stream>>>(
            in, p, mask, out, (unsigned int)n);
    } else {
        note_dropout_kernel<long long><<<grid, block, 0, stream>>>(
            in, p, mask, out, n);
    }
}